// HistogramLoss_83571473646250
// MI455X (gfx1250) — compile-verified
//
#include <hip/hip_runtime.h>

#define BINS 256
#define TPB 256
#define WAVES_PER_BLOCK (TPB / 32)
#define MAX_BLOCKS 4096

// ---------------------------------------------------------------------------
// Kernel 1: zero the 2*256 u32 global histogram bins in workspace.
// ---------------------------------------------------------------------------
__global__ void hist_zero_kernel(unsigned* __restrict__ g) {
    int i = blockIdx.x * blockDim.x + threadIdx.x;
    if (i < 2 * BINS) g[i] = 0u;
}

// ---------------------------------------------------------------------------
// Binning exactly mirrors the reference:
//   idx   = clip(floor((x - 0) * 256), 0, 255)      (all in fp32)
//   valid = (x >= 0) && (x <= 1)
// ---------------------------------------------------------------------------
__device__ __forceinline__ void bin_one(float x, unsigned* __restrict__ h) {
    if (x >= 0.0f && x <= 1.0f) {
        int b = (int)floorf(x * 256.0f);
        b = (b < 0) ? 0 : ((b > BINS - 1) ? (BINS - 1) : b);
        atomicAdd(&h[b], 1u);   // ds_add_u32 on LDS
    }
}

// ---------------------------------------------------------------------------
// Kernel 2: bandwidth-bound streaming histogram, CDNA5 async data path.
//   - bulk data streamed with GLOBAL_LOAD_ASYNC_TO_LDS_B128 (ASYNCcnt,
//     double-buffered 2 x 4 KB LDS stage, s_wait_asynccnt pipelining):
//     no VGPR return data -> low register pressure -> deep occupancy
//   - global_prefetch_b8 four chunks ahead pre-stages GL2
//   - per-wave private LDS histograms (8 x 256 x u32 = 8 KB) -> low-conflict
//     ds atomics (wave32, 64 LDS banks)
//   - one global atomic add per bin per block at the end
//   LDS: 8 KB hist + 8 KB stage = 16 KB of the 320 KB WGP pool.
// ---------------------------------------------------------------------------
__global__ void __launch_bounds__(TPB)
hist_kernel(const float* __restrict__ img, unsigned* __restrict__ ghist,
            long long n) {
    __shared__ unsigned lh[WAVES_PER_BLOCK][BINS];
    __shared__ float4   stage[2][TPB];

    const int tid  = threadIdx.x;
    const int wave = tid >> 5;          // wave32 on gfx1250

#pragma unroll
    for (int w = 0; w < WAVES_PER_BLOCK; ++w) lh[w][tid] = 0u;
    __syncthreads();

    unsigned* __restrict__ h = lh[wave];

    const long long nv         = n >> 2;                     // # of float4
    const long long gstride    = (long long)gridDim.x * TPB;
    const long long blockStart = (long long)blockIdx.x * TPB;
    const float4* __restrict__ v = (const float4*)img;

    // Number of iterations for which EVERY thread in the block is in range
    // (uniform across the block -> EXEC all-ones in the async pipeline, as
    // required for well-defined ASYNCcnt waits).
    long long K = 0;
    if (nv >= blockStart + TPB) K = (nv - blockStart - TPB) / gstride + 1;

    // LDS byte addresses of this lane's staging slots (flat addr low 32 bits
    // == LDS offset per the gfx1250 LDS aperture mapping).
    const unsigned lds_a = (unsigned)(size_t)&stage[0][tid];
    const unsigned lds_b = (unsigned)(size_t)&stage[1][tid];

    const float4* p = v + blockStart + tid;

    if (K > 0) {
        asm volatile("global_load_async_to_lds_b128 %0, %1, off"
                     :: "v"(lds_a), "v"(p) : "memory");
    }
    for (long long k = 0; k < K; ++k) {
        const float4* pn = p + gstride;
        if (k + 4 < K) __builtin_prefetch(p + 4 * gstride, 0, 1); // global_prefetch_b8
        if (k + 1 < K) {
            // Issue next chunk into the other buffer, then wait for current.
            asm volatile("global_load_async_to_lds_b128 %0, %1, off"
                         :: "v"((k & 1) ? lds_a : lds_b), "v"(pn) : "memory");
            asm volatile("s_wait_asynccnt 0x1" ::: "memory");
        } else {
            asm volatile("s_wait_asynccnt 0x0" ::: "memory");
        }
        float4 x = stage[k & 1][tid];   // ds_load_b128 from staged chunk
        bin_one(x.x, h);
        bin_one(x.y, h);
        bin_one(x.z, h);
        bin_one(x.w, h);
        p = pn;
    }

    // Partial (divergent) last vec4 iteration: direct load, no async ops.
    {
        long long i = blockStart + tid + K * gstride;
        if (i < nv) {
            float4 x = v[i];
            bin_one(x.x, h);
            bin_one(x.y, h);
            bin_one(x.z, h);
            bin_one(x.w, h);
        }
    }

    // Scalar tail (n not a multiple of 4).
    for (long long i = (nv << 2) + blockStart + tid; i < n; i += gstride) {
        bin_one(img[i], h);
    }

    __syncthreads();

    // Merge the 8 per-wave histograms; single global atomic per bin per block.
    unsigned total = 0;
#pragma unroll
    for (int w = 0; w < WAVES_PER_BLOCK; ++w) total += lh[w][tid];
    if (total) atomicAdd(&ghist[tid], total);
}

// ---------------------------------------------------------------------------
// Kernel 3: normalize both histograms (exact integer totals), L1 distance,
// divide by BINS. One block of 256 threads; deterministic tree reductions.
// ---------------------------------------------------------------------------
__global__ void __launch_bounds__(BINS)
hist_loss_kernel(const unsigned* __restrict__ g, float* __restrict__ out) {
    __shared__ unsigned s1[BINS];
    __shared__ unsigned s2[BINS];
    __shared__ float    sd[BINS];

    const int t = threadIdx.x;
    const unsigned c1 = g[t];
    const unsigned c2 = g[BINS + t];
    s1[t] = c1;
    s2[t] = c2;
    __syncthreads();

    for (int off = BINS / 2; off > 0; off >>= 1) {
        if (t < off) {
            s1[t] += s1[t + off];
            s2[t] += s2[t + off];
        }
        __syncthreads();
    }

    const float t1 = (float)s1[0];
    const float t2 = (float)s2[0];
    const float h1 = (t1 > 0.0f) ? ((float)c1 / t1) : 0.0f;
    const float h2 = (t2 > 0.0f) ? ((float)c2 / t2) : 0.0f;
    sd[t] = fabsf(h1 - h2);
    __syncthreads();

    for (int off = BINS / 2; off > 0; off >>= 1) {
        if (t < off) sd[t] += sd[t + off];
        __syncthreads();
    }

    if (t == 0) out[0] = sd[0] / (float)BINS;
}

// ---------------------------------------------------------------------------
// Launch: zero bins -> histogram img1 -> histogram img2 -> loss.
// All on `stream`, graph-capture safe, fully deterministic per call.
// ---------------------------------------------------------------------------
extern "C" void kernel_launch(void* const* d_in, const int* in_sizes, int n_in,
                              void* d_out, int out_size, void* d_ws,
                              size_t ws_size, hipStream_t stream) {
    const float* img1 = (const float*)d_in[0];
    const float* img2 = (const float*)d_in[1];
    const long long n1 = (long long)in_sizes[0];
    const long long n2 = (long long)in_sizes[1];

    unsigned* g = (unsigned*)d_ws;   // [0..255]=hist1, [256..511]=hist2
    float* out  = (float*)d_out;

    hist_zero_kernel<<<2, 256, 0, stream>>>(g);

    const long long nv1 = (n1 + 3) >> 2;
    const long long nv2 = (n2 + 3) >> 2;
    long long b1 = (nv1 + TPB - 1) / TPB;
    long long b2 = (nv2 + TPB - 1) / TPB;
    if (b1 > MAX_BLOCKS) b1 = MAX_BLOCKS;
    if (b2 > MAX_BLOCKS) b2 = MAX_BLOCKS;
    if (b1 < 1) b1 = 1;
    if (b2 < 1) b2 = 1;

    hist_kernel<<<(int)b1, TPB, 0, stream>>>(img1, g, n1);
    hist_kernel<<<(int)b2, TPB, 0, stream>>>(img2, g + BINS, n2);

    hist_loss_kernel<<<1, BINS, 0, stream>>>(g, out);
}